// CrossShift_77275051589917
// MI455X (gfx1250) — compile-verified
//
#include <hip/hip_runtime.h>
#include <hip/hip_bf16.h>
#include <stdint.h>

// CrossShift: x[16,256,256,64] f32 -> out[16,257,257,64] f32 with zero row/col
// inserted at index 128 of H and W. Pure data movement, HBM-bandwidth bound
// (~540 MB total -> ~23 us at 23.3 TB/s). Implemented with gfx1250 async
// Global<->LDS DMA (ASYNCcnt path): per-lane B128 transfers, one LDS staging
// buffer per block, single s_wait_asynccnt between the load and store phases.

#define BATCH 16
#define HIN   256
#define WIN   256
#define CH    64
#define HOUT  257
#define WOUT  257

typedef float f4 __attribute__((ext_vector_type(4)));

// Async copy helpers: the ISA adds INST_OFFSET to BOTH the LDS address and the
// global address, and our LDS/global strides are identical (4096 B per
// iteration = 256 lanes * 16 B), so one base VGPR pair + one LDS-offset VGPR
// covers a whole 32 KB segment via offset immediates.
#define A_LOAD(ldsoff, gaddr, off)                                            \
    asm volatile("global_load_async_to_lds_b128 %0, %1, off offset:" #off     \
                 :: "v"(ldsoff), "v"(gaddr) : "memory")

#define A_STORE(gaddr, ldsoff, off)                                           \
    asm volatile("global_store_async_from_lds_b128 %0, %1, off offset:" #off  \
                 :: "v"(gaddr), "v"(ldsoff) : "memory")

#define A_LOAD_SEG(ldsoff, gaddr)   \
    A_LOAD(ldsoff, gaddr, 0);       \
    A_LOAD(ldsoff, gaddr, 4096);    \
    A_LOAD(ldsoff, gaddr, 8192);    \
    A_LOAD(ldsoff, gaddr, 12288);   \
    A_LOAD(ldsoff, gaddr, 16384);   \
    A_LOAD(ldsoff, gaddr, 20480);   \
    A_LOAD(ldsoff, gaddr, 24576);   \
    A_LOAD(ldsoff, gaddr, 28672)

#define A_STORE_SEG(gaddr, ldsoff)  \
    A_STORE(gaddr, ldsoff, 0);      \
    A_STORE(gaddr, ldsoff, 4096);   \
    A_STORE(gaddr, ldsoff, 8192);   \
    A_STORE(gaddr, ldsoff, 12288);  \
    A_STORE(gaddr, ldsoff, 16384);  \
    A_STORE(gaddr, ldsoff, 20480);  \
    A_STORE(gaddr, ldsoff, 24576);  \
    A_STORE(gaddr, ldsoff, 28672)

__global__ __launch_bounds__(256) void CrossShift_kernel(
        const float* __restrict__ in, float* __restrict__ out) {
    const int tid = threadIdx.x;       // 0..255
    const int hp  = blockIdx.x;        // output row 0..256
    const int b   = blockIdx.y;        // batch 0..15

    float* outRow = out + ((size_t)b * HOUT + hp) * (size_t)(WOUT * CH);

    if (hp == 128) {
        // Zero row: 257*64 floats = 4112 float4 stores across 256 threads.
        const f4 z = {0.f, 0.f, 0.f, 0.f};
        f4* o4 = (f4*)outRow;
        for (int i = tid; i < (WOUT * CH) / 4; i += 256) o4[i] = z;
        return;
    }

    const int h = hp - (hp > 128 ? 1 : 0);
    const float* inRow = in + ((size_t)b * HIN + h) * (size_t)(WIN * CH);

    // Per-thread bases: lane stride 16 B, iteration stride 4096 B.
    // Dynamic LDS only -> segment 0 at LDS byte 0, segment 1 at 32768.
    const unsigned lds0 = (unsigned)(tid * 16);
    const unsigned lds1 = 32768u + (unsigned)(tid * 16);

    const uint64_t src0 = (uint64_t)(uintptr_t)inRow + (uint64_t)(tid * 16);
    const uint64_t src1 = (uint64_t)(uintptr_t)(inRow + 128 * CH) + (uint64_t)(tid * 16);
    const uint64_t dst0 = (uint64_t)(uintptr_t)outRow + (uint64_t)(tid * 16);
    const uint64_t dst1 = (uint64_t)(uintptr_t)(outRow + 129 * CH) + (uint64_t)(tid * 16);

    // Phase 1: 16 async B128 loads in flight (2 x 32 KB segments -> LDS).
    A_LOAD_SEG(lds0, src0);
    A_LOAD_SEG(lds1, src1);

    // Each lane stores exactly the LDS bytes it loaded, so a per-wave
    // asynccnt drain is sufficient (no workgroup barrier needed).
    asm volatile("s_wait_asynccnt 0" ::: "memory");

    // Phase 2: 16 async B128 stores (LDS -> shifted output segments).
    A_STORE_SEG(dst0, lds0);
    A_STORE_SEG(dst1, lds1);

    // Zero the center pixel w'=128 (64 floats = 16 float4).
    if (tid < 16) {
        const f4 z = {0.f, 0.f, 0.f, 0.f};
        ((f4*)(outRow + 128 * CH))[tid] = z;
    }

    // Drain outstanding async stores (S_ENDPGM also implies wait-idle).
    asm volatile("s_wait_asynccnt 0" ::: "memory");
}

extern "C" void kernel_launch(void* const* d_in, const int* in_sizes, int n_in,
                              void* d_out, int out_size, void* d_ws, size_t ws_size,
                              hipStream_t stream) {
    (void)in_sizes; (void)n_in; (void)out_size; (void)d_ws; (void)ws_size;
    const float* x = (const float*)d_in[0];
    float* out = (float*)d_out;

    dim3 grid(HOUT, BATCH, 1);   // one block per output row (b, h')
    dim3 block(256, 1, 1);       // 8 wave32s
    size_t ldsBytes = 65536;     // 2 x 32 KB staging segments (dynamic LDS @ offset 0)
    CrossShift_kernel<<<grid, block, ldsBytes, stream>>>(x, out);
}